// CRN_Discriminator_8177617731799
// MI455X (gfx1250) — compile-verified
//
#include <hip/hip_runtime.h>
#include <hip/hip_bf16.h>

typedef __attribute__((ext_vector_type(2))) float v2f;
typedef __attribute__((ext_vector_type(8))) float v8f;

#define LTOT 8192          // B*M = 16*512
#define BATCH 16
#define NPTS 4096
#define MCTR 512
#define EPSBN 1e-5f

// ---------------------------------------------------------------------------
// Farthest point sampling: one block per batch, coords + min-dists in regs.
// ---------------------------------------------------------------------------
__global__ __launch_bounds__(512) void fps_kernel(const float* __restrict__ xyz,
                                                  float* __restrict__ centers) {
    const int T = 512, PPT = NPTS / T;           // 8 points per thread
    int b = blockIdx.x, t = threadIdx.x;
    const float* base = xyz + (size_t)b * NPTS * 3;

    float px[PPT], py[PPT], pz[PPT], mind[PPT];
    #pragma unroll
    for (int i = 0; i < PPT; ++i) {
        int n = t + i * T;
        px[i] = base[n * 3 + 0];
        py[i] = base[n * 3 + 1];
        pz[i] = base[n * 3 + 2];
        mind[i] = __builtin_inff();
    }

    __shared__ float sval[512];
    __shared__ int   sidx[512];
    __shared__ float cpt[3];

    int cur = 0;
    for (int s = 0; s < MCTR; ++s) {
        if (t == 0) {
            float cx = base[cur * 3 + 0], cy = base[cur * 3 + 1], cz = base[cur * 3 + 2];
            cpt[0] = cx; cpt[1] = cy; cpt[2] = cz;
            size_t o = ((size_t)b * MCTR + s) * 3;
            centers[o + 0] = cx; centers[o + 1] = cy; centers[o + 2] = cz;
        }
        __syncthreads();
        float cx = cpt[0], cy = cpt[1], cz = cpt[2];

        float bestv = -1.0f; int besti = NPTS;
        #pragma unroll
        for (int i = 0; i < PPT; ++i) {
            float dx = px[i] - cx, dy = py[i] - cy, dz = pz[i] - cz;
            float d = dx * dx + dy * dy + dz * dz;
            mind[i] = fminf(mind[i], d);
            int n = t + i * T;                     // ascending -> first-max kept
            if (mind[i] > bestv) { bestv = mind[i]; besti = n; }
        }
        sval[t] = bestv; sidx[t] = besti;
        __syncthreads();
        for (int ofs = 256; ofs > 0; ofs >>= 1) {
            if (t < ofs) {
                float v2 = sval[t + ofs]; int i2 = sidx[t + ofs];
                if (v2 > sval[t] || (v2 == sval[t] && i2 < sidx[t])) {
                    sval[t] = v2; sidx[t] = i2;
                }
            }
            __syncthreads();
        }
        cur = sidx[0];
    }
}

// ---------------------------------------------------------------------------
// Ball query: one wave32 per (b,m) row; in-order rank via ballot prefix scan.
// Writes X tensors channel-major: X[k][b*M+m].
// ---------------------------------------------------------------------------
__global__ void ball_query_kernel(const float* __restrict__ xyz,
                                  const float* __restrict__ centers,
                                  float* __restrict__ X16,
                                  float* __restrict__ X32,
                                  float* __restrict__ X128) {
    int gid = blockIdx.x * blockDim.x + threadIdx.x;
    int wave = gid >> 5, lane = gid & 31;
    if (wave >= BATCH * MCTR) return;
    int b = wave / MCTR, m = wave % MCTR;
    int col = b * MCTR + m;

    size_t co = ((size_t)b * MCTR + m) * 3;
    float cx = centers[co + 0], cy = centers[co + 1], cz = centers[co + 2];
    float cn2 = cx * cx + cy * cy + cz * cz;

    // zero-pad outputs (reference zero-fills missing neighbors)
    if (lane < 16) X16[(size_t)lane * LTOT + col] = 0.0f;
    X32[(size_t)lane * LTOT + col] = 0.0f;
    #pragma unroll
    for (int r = 0; r < 4; ++r) X128[(size_t)(lane + 32 * r) * LTOT + col] = 0.0f;

    const float r0 = 0.1f * 0.1f, r1 = 0.2f * 0.2f, r2 = 0.4f * 0.4f;
    int cnt0 = 0, cnt1 = 0, cnt2 = 0;
    unsigned mlt = (1u << lane) - 1u;

    const float* base = xyz + (size_t)b * NPTS * 3;
    for (int n0 = 0; n0 < NPTS; n0 += 32) {
        int n = n0 + lane;
        float qx = base[n * 3 + 0], qy = base[n * 3 + 1], qz = base[n * 3 + 2];
        float xn2 = qx * qx + qy * qy + qz * qz;
        float d2 = cn2 + xn2 - 2.0f * (cx * qx + cy * qy + cz * qz);
        d2 = fmaxf(d2, 0.0f);

        unsigned b0 = __builtin_amdgcn_ballot_w32(d2 < r0);
        int rk = cnt0 + __builtin_popcount(b0 & mlt);
        if ((d2 < r0) && rk < 16)  X16[(size_t)rk * LTOT + col] = d2;
        cnt0 += __builtin_popcount(b0);

        unsigned b1 = __builtin_amdgcn_ballot_w32(d2 < r1);
        rk = cnt1 + __builtin_popcount(b1 & mlt);
        if ((d2 < r1) && rk < 32)  X32[(size_t)rk * LTOT + col] = d2;
        cnt1 += __builtin_popcount(b1);

        unsigned b2 = __builtin_amdgcn_ballot_w32(d2 < r2);
        rk = cnt2 + __builtin_popcount(b2 & mlt);
        if ((d2 < r2) && rk < 128) X128[(size_t)rk * LTOT + col] = d2;
        cnt2 += __builtin_popcount(b2);
    }
}

// ---------------------------------------------------------------------------
// GEMM via V_WMMA_F32_16X16X4_F32: one wave computes a 16(O) x 64(L) strip
// of Y = W * X using 4 independent accumulator tiles (A fragment shared).
// W: [O,C] row-major, X: [C,LTOT] row-major, Y: [O,LTOT] row-major.
// ---------------------------------------------------------------------------
__global__ void gemm_wmma_f32(const float* __restrict__ W,
                              const float* __restrict__ X,
                              float* __restrict__ Y,
                              int O, int C) {
    int gid  = blockIdx.x * blockDim.x + threadIdx.x;
    int wave = gid >> 5, lane = gid & 31;
    const int tilesL = LTOT / 64;                 // 128 strips of 64 columns
    int tiles = (O >> 4) * tilesL;
    if (wave >= tiles) return;                    // wave-uniform: EXEC all-ones inside
    int to = wave / tilesL;                       // tile row (channels)
    int tl = wave % tilesL;                       // 64-col strip (points)
    int half = lane >> 4;                         // 0: K+0/1, 1: K+2/3
    int hl   = lane & 15;

    const float* wrow = W + (size_t)(to * 16 + hl) * C;
    int col0 = tl * 64 + hl;

    v8f acc0 = {}, acc1 = {}, acc2 = {}, acc3 = {};
    for (int k = 0; k < C; k += 4) {
        int ka = k + half * 2;
        v2f a;
        a.x = wrow[ka];
        a.y = wrow[ka + 1];
        const float* x0 = X + (size_t)ka * LTOT + col0;
        const float* x1 = x0 + LTOT;
        if (k + 8 < C) {                           // prefetch 1 K-step ahead
            __builtin_prefetch(x0 + 4 * LTOT, 0, 1);
            __builtin_prefetch(x1 + 4 * LTOT, 0, 1);
        }
        v2f b0 = { x0[0],  x1[0]  };
        v2f b1 = { x0[16], x1[16] };
        v2f b2 = { x0[32], x1[32] };
        v2f b3 = { x0[48], x1[48] };
        acc0 = __builtin_amdgcn_wmma_f32_16x16x4_f32(false, a, false, b0, (short)0, acc0, false, false);
        acc1 = __builtin_amdgcn_wmma_f32_16x16x4_f32(false, a, false, b1, (short)0, acc1, false, false);
        acc2 = __builtin_amdgcn_wmma_f32_16x16x4_f32(false, a, false, b2, (short)0, acc2, false, false);
        acc3 = __builtin_amdgcn_wmma_f32_16x16x4_f32(false, a, false, b3, (short)0, acc3, false, false);
    }
    #pragma unroll
    for (int j = 0; j < 8; ++j) {
        float* yr = Y + (size_t)(to * 16 + j + 8 * half) * LTOT + col0;
        yr[0]  = acc0[j];
        yr[16] = acc1[j];
        yr[32] = acc2[j];
        yr[48] = acc3[j];
    }
}

// ---------------------------------------------------------------------------
// Per-channel mean / biased variance over LTOT (BatchNorm1d training stats).
// ---------------------------------------------------------------------------
__global__ void chan_stats_kernel(const float* __restrict__ Y,
                                  float* __restrict__ mean,
                                  float* __restrict__ var) {
    int c = blockIdx.x;
    const float* row = Y + (size_t)c * LTOT;
    float s = 0.0f, s2 = 0.0f;
    for (int i = threadIdx.x; i < LTOT; i += blockDim.x) {
        float v = row[i]; s += v; s2 += v * v;
    }
    __shared__ float sh[256], sh2[256];
    sh[threadIdx.x] = s; sh2[threadIdx.x] = s2;
    __syncthreads();
    for (int ofs = 128; ofs > 0; ofs >>= 1) {
        if ((int)threadIdx.x < ofs) {
            sh[threadIdx.x]  += sh[threadIdx.x + ofs];
            sh2[threadIdx.x] += sh2[threadIdx.x + ofs];
        }
        __syncthreads();
    }
    if (threadIdx.x == 0) {
        float m = sh[0] / (float)LTOT;
        mean[c] = m;
        var[c]  = sh2[0] / (float)LTOT - m * m;
    }
}

// ---------------------------------------------------------------------------
// BN normalize + (Leaky)ReLU, in place.
// ---------------------------------------------------------------------------
__global__ void bn_act_kernel(float* __restrict__ Y,
                              const float* __restrict__ mean,
                              const float* __restrict__ var,
                              const float* __restrict__ g,
                              const float* __restrict__ b,
                              float slope, int total) {
    int i = blockIdx.x * blockDim.x + threadIdx.x;
    if (i >= total) return;
    int c = i >> 13;                               // / LTOT (8192)
    float y = Y[i];
    y = (y - mean[c]) * rsqrtf(var[c] + EPSBN) * g[c] + b[c];
    Y[i] = (y > 0.0f) ? y : slope * y;
}

// ---------------------------------------------------------------------------
// reshape(B,M,6,128).max(-1) on the flat per-batch [768,M] tensor.
// Y3 stored as [768, LTOT] with L-index = b*M+m.
// ---------------------------------------------------------------------------
__global__ void maxpool6_kernel(const float* __restrict__ Y3,
                                float* __restrict__ O18, int chOfs) {
    int i = blockIdx.x * blockDim.x + threadIdx.x;
    if (i >= BATCH * MCTR * 6) return;
    int s  = i % 6;
    int m2 = (i / 6) % MCTR;
    int b  = i / (6 * MCTR);
    int f0 = m2 * 768 + s * 128;
    float best = -3.402823466e38f;
    #pragma unroll 4
    for (int t = 0; t < 128; ++t) {
        int f = f0 + t;
        int c = f >> 9;           // / M (M=512)
        int m = f & 511;          // % M
        best = fmaxf(best, Y3[(size_t)c * LTOT + b * MCTR + m]);
    }
    O18[(size_t)(chOfs + s) * LTOT + b * MCTR + m2] = best;
}

// ---------------------------------------------------------------------------
// mlp4: 18 -> 1 dot product per point (C=18 not WMMA-friendly).
// ---------------------------------------------------------------------------
__global__ void head_dot_kernel(const float* __restrict__ O18,
                                const float* __restrict__ w,
                                float* __restrict__ y) {
    int i = blockIdx.x * blockDim.x + threadIdx.x;
    if (i >= LTOT) return;
    float s = 0.0f;
    #pragma unroll
    for (int c = 0; c < 18; ++c) s += w[c] * O18[(size_t)c * LTOT + i];
    y[i] = s;
}

// ---------------------------------------------------------------------------
// Final mean over M per batch -> d_out[B]
// ---------------------------------------------------------------------------
__global__ void final_mean_kernel(const float* __restrict__ y,
                                  float* __restrict__ out) {
    int b = blockIdx.x;
    __shared__ float sh[256];
    float s = 0.0f;
    for (int m = threadIdx.x; m < MCTR; m += blockDim.x) s += y[b * MCTR + m];
    sh[threadIdx.x] = s;
    __syncthreads();
    for (int ofs = 128; ofs > 0; ofs >>= 1) {
        if ((int)threadIdx.x < ofs) sh[threadIdx.x] += sh[threadIdx.x + ofs];
        __syncthreads();
    }
    if (threadIdx.x == 0) out[b] = sh[0] / (float)MCTR;
}

// ---------------------------------------------------------------------------
extern "C" void kernel_launch(void* const* d_in, const int* in_sizes, int n_in,
                              void* d_out, int out_size, void* d_ws, size_t ws_size,
                              hipStream_t stream) {
    (void)in_sizes; (void)n_in; (void)out_size; (void)ws_size;

    // ---- input unpack (setup_inputs dict insertion order) ----
    int idx = 0;
    const float* xyz = (const float*)d_in[idx++];

    struct Layer { const float *w, *g, *b; int O, C; };
    Layer scaleA[3][3];
    const int Ks[3] = {16, 32, 128};
    for (int s = 0; s < 3; ++s) {
        int dims[4] = {Ks[s], 32, 32, 64};
        for (int l = 0; l < 3; ++l) {
            scaleA[s][l].w = (const float*)d_in[idx++];
            scaleA[s][l].g = (const float*)d_in[idx++];
            scaleA[s][l].b = (const float*)d_in[idx++];
            scaleA[s][l].C = dims[l];
            scaleA[s][l].O = dims[l + 1];
        }
    }
    Layer mlp2[3]; { int d[4] = {64, 64, 64, 128};
        for (int l = 0; l < 3; ++l) {
            mlp2[l].w = (const float*)d_in[idx++];
            mlp2[l].g = (const float*)d_in[idx++];
            mlp2[l].b = (const float*)d_in[idx++];
            mlp2[l].C = d[l]; mlp2[l].O = d[l + 1];
        } }
    Layer mlp3[3]; { int d[4] = {128, 64, 96, 768};
        for (int l = 0; l < 3; ++l) {
            mlp3[l].w = (const float*)d_in[idx++];
            mlp3[l].g = (const float*)d_in[idx++];
            mlp3[l].b = (const float*)d_in[idx++];
            mlp3[l].C = d[l]; mlp3[l].O = d[l + 1];
        } }
    const float* w4 = (const float*)d_in[idx++];
    const float* g4 = (const float*)d_in[idx++];
    const float* b4 = (const float*)d_in[idx++];

    // ---- workspace layout (floats) ----
    float* ws = (float*)d_ws;
    const size_t L = LTOT;
    float* bufA    = ws;                       // 768*L
    float* bufB    = bufA + (size_t)768 * L;   // 768*L
    float* X16     = bufB + (size_t)768 * L;   // 16*L
    float* X32     = X16  + (size_t)16  * L;   // 32*L
    float* X128    = X32  + (size_t)32  * L;   // 128*L
    float* O18     = X128 + (size_t)128 * L;   // 18*L
    float* yv      = O18  + (size_t)18  * L;   // L
    float* centers = yv + L;                   // B*M*3
    float* meanBuf = centers + (size_t)BATCH * MCTR * 3;  // 768
    float* varBuf  = meanBuf + 768;                        // 768

    // ---- geometry ----
    fps_kernel<<<BATCH, 512, 0, stream>>>(xyz, centers);
    {
        int waves = BATCH * MCTR;
        int thr   = waves * 32;
        ball_query_kernel<<<(thr + 255) / 256, 256, 0, stream>>>(
            xyz, centers, X16, X32, X128);
    }

    // ---- layer driver: WMMA GEMM -> BN stats -> normalize+activation ----
    auto run_layer = [&](const Layer& ly, const float* Xin, float* Yout, float slope) {
        int tiles = (ly.O / 16) * (LTOT / 64);
        gemm_wmma_f32<<<(tiles + 7) / 8, 256, 0, stream>>>(
            ly.w, Xin, Yout, ly.O, ly.C);
        chan_stats_kernel<<<ly.O, 256, 0, stream>>>(Yout, meanBuf, varBuf);
        int total = ly.O * LTOT;
        bn_act_kernel<<<(total + 255) / 256, 256, 0, stream>>>(
            Yout, meanBuf, varBuf, ly.g, ly.b, slope, total);
    };

    const float* Xs[3] = {X16, X32, X128};
    for (int s = 0; s < 3; ++s) {
        run_layer(scaleA[s][0], Xs[s], bufA, 0.01f);   // K  -> 32
        run_layer(scaleA[s][1], bufA,  bufB, 0.01f);   // 32 -> 32
        run_layer(scaleA[s][2], bufB,  bufA, 0.01f);   // 32 -> 64
        run_layer(mlp2[0],      bufA,  bufB, 0.01f);   // 64 -> 64
        run_layer(mlp2[1],      bufB,  bufA, 0.01f);   // 64 -> 64
        run_layer(mlp2[2],      bufA,  bufB, 0.01f);   // 64 -> 128
        run_layer(mlp3[0],      bufB,  bufA, 0.01f);   // 128-> 64
        run_layer(mlp3[1],      bufA,  bufB, 0.01f);   // 64 -> 96
        run_layer(mlp3[2],      bufB,  bufA, 0.01f);   // 96 -> 768
        int n = BATCH * MCTR * 6;
        maxpool6_kernel<<<(n + 255) / 256, 256, 0, stream>>>(bufA, O18, 6 * s);
    }

    // ---- head: 18 -> 1, BN(1 channel) + ReLU, mean over M ----
    head_dot_kernel<<<(LTOT + 255) / 256, 256, 0, stream>>>(O18, w4, yv);
    chan_stats_kernel<<<1, 256, 0, stream>>>(yv, meanBuf, varBuf);
    bn_act_kernel<<<(LTOT + 255) / 256, 256, 0, stream>>>(
        yv, meanBuf, varBuf, g4, b4, 0.0f, LTOT);
    final_mean_kernel<<<BATCH, 256, 0, stream>>>(yv, (float*)d_out);
}